// MotionCliffordSharedSimplicialMPNN_2774548873290
// MI455X (gfx1250) — compile-verified
//
#include <hip/hip_runtime.h>
#include <hip/hip_bf16.h>

typedef float v2f __attribute__((ext_vector_type(2)));
typedef float v8f __attribute__((ext_vector_type(8)));

// ---- Clifford Cl(3,0) tables: blades {1,e0,e1,e2,e01,e02,e12,e012} ----
constexpr int GBc[8] = {0,1,1,1,2,2,2,3};
// product blade index J[a*8+b] and sign S[a*8+b] for blade_a * blade_b
constexpr int CJ[64] = {
  0,1,2,3,4,5,6,7,
  1,0,4,5,2,3,7,6,
  2,4,0,6,1,7,3,5,
  3,5,6,0,7,1,2,4,
  4,2,1,7,0,6,5,3,
  5,3,7,1,6,0,4,2,
  6,7,3,2,5,4,0,1,
  7,6,5,4,3,2,1,0};
constexpr float CS[64] = {
  1, 1, 1, 1, 1, 1, 1, 1,
  1, 1, 1, 1, 1, 1, 1, 1,
  1,-1, 1, 1,-1,-1, 1,-1,
  1,-1,-1, 1, 1,-1,-1, 1,
  1,-1, 1, 1,-1,-1, 1,-1,
  1,-1,-1, 1, 1,-1,-1, 1,
  1, 1,-1, 1,-1, 1,-1,-1,
  1, 1,-1, 1,-1, 1,-1,-1};

__device__ inline float sigm(float z){ return 1.0f/(1.0f+__expf(-z)); }
__device__ inline float qroot(float q){ return sqrtf(sqrtf(q*q + 1e-16f)); }

// f32 WMMA: D(16x16) = A(16x4) * B(4x16) + C ; keeps reference f32 numerics
__device__ inline v8f wmma4(v2f a, v2f b, v8f c){
  return __builtin_amdgcn_wmma_f32_16x16x4_f32(false, a, false, b, (short)0, c, false, false);
}

// =====================================================================
// Fused CEMLP block: y = mv_layernorm(sgp(mv_silu(mv_linear(x))))
// X:(Nrows,nin,8) -> Y:(Nrows,16,8).  One workgroup = 16 rows, 8 waves,
// wave w owns blade w for the three WMMA matmuls (lin, right, left).
// All WMMA operands come from LDS: weights are staged (zero-padded) once
// per workgroup so the inner loops are pure ds_load + v_wmma.
// Global traffic uses b128 (float4) loads/stores.
// =====================================================================
__global__ __launch_bounds__(256) void cemlp_block_kernel(
    const float* __restrict__ X, float* __restrict__ Y,
    const float* __restrict__ residual, int Nrows, int nin,
    const float* __restrict__ w_lin, const float* __restrict__ b_lin,
    const float* __restrict__ silu_a, const float* __restrict__ silu_b,
    const float* __restrict__ w_right, const float* __restrict__ norm_a,
    const float* __restrict__ gp_w,
    const float* __restrict__ w_left, const float* __restrict__ b_left,
    const float* __restrict__ ln_a)
{
  __shared__ float xs[16*36*8];    // input tile, K padded to <=36 (red aliases this)
  __shared__ float ts[16*16*8];    // after lin+silu
  __shared__ float us[16*16*8];    // after right+normalization
  __shared__ float ysm[16*16*8];   // gp, then (left+gp)/sqrt2
  __shared__ float wl_s[16*36*4];  // staged w_lin, zero-padded K
  __shared__ float wr_s[16*16*4];  // staged w_right
  __shared__ float wf_s[16*16*4];  // staged w_left
  __shared__ float gpw_s[16*4*4*4];// staged gp_w

  const int tid  = threadIdx.x;
  const int KP   = (nin + 3) & ~3;
  const int row0 = blockIdx.x * 16;

  // ---- stage weights into LDS ----
  for (int idx = tid; idx < 16*KP; idx += 256) {   // one float4 per (m,k)
    int k = idx % KP;
    int m = idx / KP;
    float4 v = {0.f,0.f,0.f,0.f};
    if (k < nin) v = *(const float4*)(w_lin + (m*nin + k)*4);
    *(float4*)(wl_s + (m*36 + k)*4) = v;
  }
  // contiguous 1024-float params: one float4 per thread
  ((float4*)wr_s)[tid]  = ((const float4*)w_right)[tid];
  ((float4*)wf_s)[tid]  = ((const float4*)w_left)[tid];
  ((float4*)gpw_s)[tid] = ((const float4*)gp_w)[tid];

  // ---- stage input tile into LDS (zero-padded K), b128 chunks ----
  for (int idx = tid; idx < 16*KP*2; idx += 256) {
    int hf = idx & 1;                 // which half of the 8 blades
    int k  = (idx >> 1) % KP;
    int r  = idx / (KP*2);
    int rg = row0 + r;
    float4 v = {0.f,0.f,0.f,0.f};
    if (k < nin && rg < Nrows)
      v = *(const float4*)(X + ((long)rg*nin + k)*8 + hf*4);
    *(float4*)(xs + (r*36 + k)*8 + hf*4) = v;
  }
  __syncthreads();

  const int blade = tid >> 5;       // wave id == blade
  const int lane  = tid & 31;
  const int half  = lane >> 4;
  const int l16   = lane & 15;
  const int g     = GBc[blade];
  const int prow  = tid >> 4;       // pointwise (row,chan) mapping
  const int pcol  = tid & 15;

  // operand fetch helpers (all from LDS, branch-free)
  auto mka = [&](const float* wlds, int stride, int k0)->v2f {
    int ka = k0 + 2*half;
    v2f a; a.x = wlds[(l16*stride + ka)*4 + g];
           a.y = wlds[(l16*stride + ka+1)*4 + g];
    return a;
  };
  auto mkb = [&](const float* xlds, int stride, int k0)->v2f {
    int ka = k0 + 2*half;
    v2f b; b.x = xlds[(l16*stride + ka)*8 + blade];
           b.y = xlds[(l16*stride + ka+1)*8 + blade];
    return b;
  };
  const v8f vz = {0.f,0.f,0.f,0.f,0.f,0.f,0.f,0.f};

  // ---------------- LIN: t = W_lin[:, :, g] @ X_blade ----------------
  {
    v8f acc0 = vz, acc1 = vz;
    int k0 = 0;
    for (; k0 + 8 <= KP; k0 += 8) {                 // two independent chains
      acc0 = wmma4(mka(wl_s,36,k0  ), mkb(xs,36,k0  ), acc0);
      acc1 = wmma4(mka(wl_s,36,k0+4), mkb(xs,36,k0+4), acc1);
    }
    if (k0 < KP)
      acc0 = wmma4(mka(wl_s,36,k0), mkb(xs,36,k0), acc0);
#pragma unroll
    for (int r = 0; r < 8; ++r) {
      int chan = r + 8*half;
      float v = acc0[r] + acc1[r];
      if (blade == 0) v += b_lin[chan];     // bias embeds on grade-0 blade only
      ts[(l16*16 + chan)*8 + blade] = v;
    }
  }
  __syncthreads();

  // ---------------- mv_silu (per row,chan across blades) ----------------
  {
    float v[8];
#pragma unroll
    for (int i=0;i<8;++i) v[i] = ts[(prow*16+pcol)*8+i];
    float invs[4];
    invs[0] = v[0];
    invs[1] = v[1]*v[1]+v[2]*v[2]+v[3]*v[3];
    invs[2] = v[4]*v[4]+v[5]*v[5]+v[6]*v[6];
    invs[3] = v[7]*v[7];
    float gate[4];
#pragma unroll
    for (int gg=0; gg<4; ++gg)
      gate[gg] = sigm(silu_a[pcol*4+gg]*invs[gg] + silu_b[pcol*4+gg]);
#pragma unroll
    for (int i=0;i<8;++i) ts[(prow*16+pcol)*8+i] = v[i]*gate[GBc[i]];
  }
  __syncthreads();

  // ---------------- RIGHT: u = W_right @ t ----------------
  {
    v8f acc0 = vz, acc1 = vz;
    acc0 = wmma4(mka(wr_s,16,0 ), mkb(ts,16,0 ), acc0);
    acc1 = wmma4(mka(wr_s,16,4 ), mkb(ts,16,4 ), acc1);
    acc0 = wmma4(mka(wr_s,16,8 ), mkb(ts,16,8 ), acc0);
    acc1 = wmma4(mka(wr_s,16,12), mkb(ts,16,12), acc1);
#pragma unroll
    for (int r = 0; r < 8; ++r)
      us[(l16*16 + (r+8*half))*8 + blade] = acc0[r] + acc1[r];
  }
  __syncthreads();

  // ---------------- normalization on u ----------------
  {
    float v[8];
#pragma unroll
    for (int i=0;i<8;++i) v[i] = us[(prow*16+pcol)*8+i];
    float q[4];
    q[0]=v[0]*v[0];
    q[1]=v[1]*v[1]+v[2]*v[2]+v[3]*v[3];
    q[2]=v[4]*v[4]+v[5]*v[5]+v[6]*v[6];
    q[3]=v[7]*v[7];
    float dn[4];
#pragma unroll
    for (int gg=0; gg<4; ++gg){
      float nr = qroot(q[gg]);
      dn[gg] = sigm(norm_a[pcol*4+gg])*(nr-1.0f) + 1.0f + 1e-6f;
    }
#pragma unroll
    for (int i=0;i<8;++i) us[(prow*16+pcol)*8+i] = v[i]/dn[GBc[i]];
  }
  __syncthreads();

  // ---------------- geometric product gp_j = sum s*w*t_i*u_k ----------------
  {
    float tv[8], uv[8], gp[8];
#pragma unroll
    for (int i=0;i<8;++i){
      tv[i]=ts[(prow*16+pcol)*8+i];
      uv[i]=us[(prow*16+pcol)*8+i];
      gp[i]=0.f;
    }
#pragma unroll
    for (int i=0;i<8;++i){
#pragma unroll
      for (int k=0;k<8;++k){
        const int j = CJ[i*8+k];
        gp[j] += CS[i*8+k] * gpw_s[((pcol*4+GBc[i])*4+GBc[j])*4+GBc[k]] * tv[i]*uv[k];
      }
    }
#pragma unroll
    for (int i=0;i<8;++i) ysm[(prow*16+pcol)*8+i] = gp[i];
  }
  __syncthreads();

  // ---------------- LEFT: y = (W_left@t + b + gp)/sqrt(2) ----------------
  {
    v8f acc0 = vz, acc1 = vz;
    acc0 = wmma4(mka(wf_s,16,0 ), mkb(ts,16,0 ), acc0);
    acc1 = wmma4(mka(wf_s,16,4 ), mkb(ts,16,4 ), acc1);
    acc0 = wmma4(mka(wf_s,16,8 ), mkb(ts,16,8 ), acc0);
    acc1 = wmma4(mka(wf_s,16,12), mkb(ts,16,12), acc1);
    const float rs2 = 0.7071067811865476f;
#pragma unroll
    for (int r = 0; r < 8; ++r) {
      int chan = r + 8*half;
      float v = acc0[r] + acc1[r];
      if (blade == 0) v += b_left[chan];
      int o = (l16*16 + chan)*8 + blade;
      ysm[o] = (ysm[o] + v) * rs2;
    }
  }
  __syncthreads();

  // ---------------- mv_layernorm + optional residual + b128 store ----------------
  {
    float* red = xs;   // xs is dead after LIN; reuse as 16x16 reduction buffer
    float v[8];
#pragma unroll
    for (int i=0;i<8;++i) v[i] = ysm[(prow*16+pcol)*8+i];
    float q = 0.f;
#pragma unroll
    for (int i=0;i<8;++i) q += v[i]*v[i];
    red[prow*16+pcol] = qroot(q);
    __syncthreads();
    float m = 0.f;
#pragma unroll
    for (int cc=0;cc<16;++cc) m += red[prow*16+cc];
    m = m*(1.0f/16.0f) + 1e-6f;
    float al = ln_a[pcol];
    int rg = row0 + prow;
    if (rg < Nrows){
      long base = ((long)rg*16 + pcol)*8;
      float o[8];
#pragma unroll
      for (int i=0;i<8;++i) o[i] = al*v[i]/m;
      if (residual){
        float4 r0 = *(const float4*)(residual + base);
        float4 r1 = *(const float4*)(residual + base + 4);
        o[0]+=r0.x; o[1]+=r0.y; o[2]+=r0.z; o[3]+=r0.w;
        o[4]+=r1.x; o[5]+=r1.y; o[6]+=r1.z; o[7]+=r1.w;
      }
      float4 s0 = {o[0],o[1],o[2],o[3]};
      float4 s1 = {o[4],o[5],o[6],o[7]};
      *(float4*)(Y + base)     = s0;
      *(float4*)(Y + base + 4) = s1;
    }
  }
}

// =====================================================================
// Auxiliary kernels (b128 global traffic where possible)
// =====================================================================
__global__ void copy4_kernel(float4* d, const float4* s, long n4){
  long t = (long)blockIdx.x*256 + threadIdx.x;
  if (t < n4) d[t] = s[t];
}
__global__ void zero4_kernel(float4* p, long n4){
  long t = (long)blockIdx.x*256 + threadIdx.x;
  float4 z = {0.f,0.f,0.f,0.f};
  if (t < n4) p[t] = z;
}
// subtract per-graph mean over the N0G grade-0 nodes of each graph
__global__ void center_kernel(float* pos_c, const int* idx0, int n0g){
  __shared__ float s[3][256];
  int b = blockIdx.x, t = threadIdx.x;
  float a0=0.f,a1=0.f,a2=0.f;
  for (int j=t;j<n0g;j+=256){
    int id = idx0[b*n0g+j];
    a0 += pos_c[id*3+0]; a1 += pos_c[id*3+1]; a2 += pos_c[id*3+2];
  }
  s[0][t]=a0; s[1][t]=a1; s[2][t]=a2;
  __syncthreads();
  for (int o=128;o>0;o>>=1){
    if (t<o){ s[0][t]+=s[0][t+o]; s[1][t]+=s[1][t+o]; s[2][t]+=s[2][t+o]; }
    __syncthreads();
  }
  float m0=s[0][0]/n0g, m1=s[1][0]/n0g, m2=s[2][0]/n0g;
  for (int j=t;j<n0g;j+=256){
    int id = idx0[b*n0g+j];
    pos_c[id*3+0]-=m0; pos_c[id*3+1]-=m1; pos_c[id*3+2]-=m2;
  }
}
__global__ void natt_kernel(float* natt, const int* nt, const float* sim, int N){
  int t = blockIdx.x*256 + threadIdx.x;
  if (t >= N*3) return;
  natt[t] = sim[nt[t/3]*3 + (t%3)];
}
// grade-0 embedding: x[node] = cl0(pos,vel) (dense 16x2, bias on blade0)
__global__ void embed0_kernel(float* x, const float* pos_c, const float* vel,
    const int* idx0, const int* x_ind, const int* x_ind_ptr, const int* x_ind_batch,
    const float* w, const float* b, int n0){
  int t = blockIdx.x*256 + threadIdx.x;
  if (t >= n0*16) return;
  int c = t & 15, j = t >> 4;
  int row  = idx0[j];
  int start = x_ind_ptr[x_ind_batch[row]];
  int node  = x_ind[row*3+0] + start;
  float w0 = w[c*2], w1 = w[c*2+1];
  float4 s0, s1 = {0.f,0.f,0.f,0.f};
  s0.x = b[c];
  s0.y = w0*pos_c[node*3+0] + w1*vel[node*3+0];
  s0.z = w0*pos_c[node*3+1] + w1*vel[node*3+1];
  s0.w = w0*pos_c[node*3+2] + w1*vel[node*3+2];
  long base = ((long)row*16 + c)*8;
  *(float4*)(x + base)     = s0;
  *(float4*)(x + base + 4) = s1;
}
// build 1-simplex features: inst = j*2+p, 4 rows (2 pos, 2 vel) grade-1 embedded
__global__ void feats1_kernel(float* F, const float* pos_c, const float* vel,
    const int* idx1, const int* x_ind, const int* x_ind_ptr, const int* x_ind_batch, int n1){
  long t = (long)blockIdx.x*256 + threadIdx.x;   // one per (inst, row)
  if (t >= (long)n1*2*4) return;
  int r = (int)(t & 3); long inst = t >> 2;
  int p = (int)(inst & 1); long j = inst >> 1;
  int sidx = idx1[j];
  int start = x_ind_ptr[x_ind_batch[sidx]];
  const int perm1[2][2] = {{0,1},{1,0}};
  int a = (r<2) ? r : r-2;
  int node = x_ind[sidx*3 + perm1[p][a]] + start;
  const float* src = (r<2) ? pos_c : vel;
  float4 s0 = {0.f, src[node*3+0], src[node*3+1], src[node*3+2]};
  float4 s1 = {0.f,0.f,0.f,0.f};
  *(float4*)(F + t*8)     = s0;
  *(float4*)(F + t*8 + 4) = s1;
}
// build 2-simplex features: inst = j*6+p, 6 rows (3 pos, 3 vel)
__global__ void feats2_kernel(float* F, const float* pos_c, const float* vel,
    const int* idx2, const int* x_ind, const int* x_ind_ptr, const int* x_ind_batch, int n2){
  long t = (long)blockIdx.x*256 + threadIdx.x;   // one per (inst, row)
  if (t >= (long)n2*6*6) return;
  int r = (int)(t % 6); long inst = t / 6;
  int p = (int)(inst % 6); long j = inst / 6;
  int sidx = idx2[j];
  int start = x_ind_ptr[x_ind_batch[sidx]];
  const int perm2[6][3] = {{0,1,2},{0,2,1},{1,0,2},{1,2,0},{2,0,1},{2,1,0}};
  int a = (r<3) ? r : r-3;
  int node = x_ind[sidx*3 + perm2[p][a]] + start;
  const float* src = (r<3) ? pos_c : vel;
  float4 s0 = {0.f, src[node*3+0], src[node*3+1], src[node*3+2]};
  float4 s1 = {0.f,0.f,0.f,0.f};
  *(float4*)(F + t*8)     = s0;
  *(float4*)(F + t*8 + 4) = s1;
}
__global__ void sum_perms_kernel(float* x, const float* src, const int* idxarr, int count, int f){
  long t = (long)blockIdx.x*256 + threadIdx.x;   // one float4 per thread
  if (t >= (long)count*32) return;
  int q4 = (int)(t & 31); long j = t >> 5;
  float4 s = {0.f,0.f,0.f,0.f};
  for (int p=0;p<f;++p){
    float4 v = *(const float4*)(src + (j*f+p)*128 + q4*4);
    s.x+=v.x; s.y+=v.y; s.z+=v.z; s.w+=v.w;
  }
  *(float4*)(x + (long)idxarr[j]*128 + q4*4) = s;
}
// m_in = concat([h[dst]-h[src], node_attr[src], node_attr[dst]])
__global__ void edge_in_kernel(float* m_in, const float* x, const float* natt,
                               const int* ei, int E){
  long t = (long)blockIdx.x*256 + threadIdx.x;   // one per (e, row)
  if (t >= (long)E*22) return;
  int r = (int)(t % 22); long e = t/22;
  int src = ei[e], dst = ei[E+e];
  float4 s0 = {0.f,0.f,0.f,0.f}, s1 = {0.f,0.f,0.f,0.f};
  if (r < 16){
    long db = (long)dst*128 + r*8, sb = (long)src*128 + r*8;
    float4 d0 = *(const float4*)(x + db),   d1 = *(const float4*)(x + db + 4);
    float4 e0 = *(const float4*)(x + sb),   e1 = *(const float4*)(x + sb + 4);
    s0.x=d0.x-e0.x; s0.y=d0.y-e0.y; s0.z=d0.z-e0.z; s0.w=d0.w-e0.w;
    s1.x=d1.x-e1.x; s1.y=d1.y-e1.y; s1.z=d1.z-e1.z; s1.w=d1.w-e1.w;
  } else if (r < 19){
    s0.x = natt[src*3 + (r-16)];
  } else {
    s0.x = natt[dst*3 + (r-19)];
  }
  *(float4*)(m_in + t*8)     = s0;
  *(float4*)(m_in + t*8 + 4) = s1;
}
__global__ void cnt_kernel(float* cnt, const int* ei, int E){
  int e = blockIdx.x*256 + threadIdx.x;
  if (e < E) atomicAdd(&cnt[ei[E+e]], 1.0f);
}
__global__ void scatter_kernel(float* agg, const float* msg, const int* ei, int E){
  long t = (long)blockIdx.x*256 + threadIdx.x;
  if (t >= (long)E*128) return;
  int q = (int)(t & 127); long e = t >> 7;
  atomicAdd(&agg[(long)ei[E+e]*128 + q], msg[t]);
}
__global__ void aggnorm_kernel(float* agg, const float* cnt, int N){
  long t = (long)blockIdx.x*256 + threadIdx.x;   // one float4 per thread
  if (t >= (long)N*32) return;
  float c = fmaxf(cnt[t>>5], 1.0f);
  float4 v = *(const float4*)(agg + t*4);
  v.x/=c; v.y/=c; v.z/=c; v.w/=c;
  *(float4*)(agg + t*4) = v;
}
// upd_in = concat([h, agg, node_attr])  (35 rows)
__global__ void node_in_kernel(float* nb, const float* x, const float* agg,
                               const float* natt, int N){
  long t = (long)blockIdx.x*256 + threadIdx.x;   // one per (n, row)
  if (t >= (long)N*35) return;
  int r = (int)(t % 35); long n = t/35;
  float4 s0 = {0.f,0.f,0.f,0.f}, s1 = {0.f,0.f,0.f,0.f};
  if (r < 16){
    s0 = *(const float4*)(x + n*128 + r*8);
    s1 = *(const float4*)(x + n*128 + r*8 + 4);
  } else if (r < 32){
    s0 = *(const float4*)(agg + n*128 + (r-16)*8);
    s1 = *(const float4*)(agg + n*128 + (r-16)*8 + 4);
  } else {
    s0.x = natt[n*3 + (r-32)];
  }
  *(float4*)(nb + t*8)     = s0;
  *(float4*)(nb + t*8 + 4) = s1;
}
__global__ void gather_nodes_kernel(float* dst, const float* x, const int* idx0, int n0){
  long t = (long)blockIdx.x*256 + threadIdx.x;   // one float4 per thread
  if (t >= (long)n0*32) return;
  int q4 = (int)(t & 31); long j = t >> 5;
  *(float4*)(dst + t*4) = *(const float4*)(x + (long)idx0[j]*128 + q4*4);
}
// pred = node_pos + proj_lin(out)[...,0,1:4]; per-node MSE loss
__global__ void predloss_kernel(float* out, const float* po, const float* plw,
    const float* pos_c, const int* idx0, const float* yref, int n0){
  int j = blockIdx.x*256 + threadIdx.x;
  if (j >= n0) return;
  int node = idx0[j];
  float loss = 0.f;
  for (int k=0;k<3;++k){
    int i = k+1;                 // grade-1 blades; GB[i]==1
    float acc = 0.f;
    for (int m=0;m<16;++m) acc += po[((long)j*16+m)*8+i] * plw[m*4 + 1];
    float pred = pos_c[node*3+k] + acc;
    float d = pred - yref[j*3+k];
    loss += d*d;
  }
  out[1+j] = loss * (1.0f/3.0f);
}
__global__ void lossmean_kernel(float* out, int n){
  __shared__ float s[256];
  int t = threadIdx.x;
  float a = 0.f;
  for (int j=t;j<n;j+=256) a += out[1+j];
  s[t] = a; __syncthreads();
  for (int o=128;o>0;o>>=1){ if (t<o) s[t]+=s[t+o]; __syncthreads(); }
  if (t==0) out[0] = s[0]/n;
}

// =====================================================================
static inline int cdiv(long a, long b){ return (int)((a + b - 1)/b); }

extern "C" void kernel_launch(void* const* d_in, const int* in_sizes, int n_in,
                              void* d_out, int out_size, void* d_ws, size_t ws_size,
                              hipStream_t stream) {
  (void)in_sizes; (void)out_size; (void)ws_size;
  const int B=8, N0G=512, N1G=1024, N2G=512, PG=2048;
  const int N = B*PG, N0 = B*N0G, N1 = B*N1G, N2 = B*N2G, E = 65536;

  const float* pos        = (const float*)d_in[0];
  const float* vel        = (const float*)d_in[1];
  const float* yref       = (const float*)d_in[2];
  const int*   x_ind      = (const int*)d_in[3];
  const int*   node_types = (const int*)d_in[4];
  const int*   edge_index = (const int*)d_in[5];
  const int*   x_ind_ptr  = (const int*)d_in[7];
  const int*   x_ind_batch= (const int*)d_in[8];
  const int*   idx0       = (const int*)d_in[9];
  const int*   idx1       = (const int*)d_in[10];
  const int*   idx2       = (const int*)d_in[11];

  // ---- params: jax pytree flatten order (alphabetical keys) starting at 12 ----
  int pi = 12;
  auto P = [&](void)->const float* {
    int k = (pi < n_in) ? pi : (n_in-1); pi++;
    return (const float*)d_in[k];
  };
  struct BlockP { const float *lin_b,*lin_w,*ln_a,*gp_w,*ll_b,*ll_w,*lr_w,*norm_a,*silu_a,*silu_b; };
  auto RB = [&](void)->BlockP {
    BlockP b;
    b.lin_b=P(); b.lin_w=P();                 // lin.b, lin.w
    b.ln_a=P();                               // ln.a
    b.gp_w=P();                               // sgp.gp_w
    b.ll_b=P(); b.ll_w=P();                   // sgp.linear_left.{b,w}
    b.lr_w=P();                               // sgp.linear_right.w
    b.norm_a=P();                             // sgp.norm.a
    b.silu_a=P(); b.silu_b=P();               // silu.{a,b}
    return b;
  };
  const float* cl0_b = P(); const float* cl0_w = P();       // cl0
  BlockP cl1   = RB();                                       // cl1 (1 block)
  BlockP cl2_0 = RB(), cl2_1 = RB();                         // cl2 (2 blocks)
  BlockP ed[4][2], nd[4][2];                                 // layers
  for (int l=0;l<4;++l){ ed[l][0]=RB(); ed[l][1]=RB(); nd[l][0]=RB(); nd[l][1]=RB(); }
  const float* pl_b = P(); const float* pl_w = P();          // proj_lin
  BlockP pm = RB();                                          // proj_mlp
  const float* sim_emb = P();                                // sim_emb
  (void)pl_b;

  // ---- workspace layout (floats; every buffer float4-aligned) ----
  float* W = (float*)d_ws; size_t off = 0;
  auto alloc = [&](size_t n){ float* p = W + off; off += n; return p; };
  float* pos_c = alloc((size_t)N*3);
  float* natt  = alloc((size_t)N*3);
  float* xA    = alloc((size_t)N*128);
  float* xB    = alloc((size_t)N*128);
  float* agg   = alloc((size_t)N*128);
  float* cnt   = alloc((size_t)N);
  float* ninb  = alloc((size_t)N*35*8);
  float* m_in  = alloc((size_t)E*22*8);
  float* msgA  = alloc((size_t)E*128);
  float* msgB  = alloc((size_t)E*128);

  auto run_block = [&](const float* Xp, float* Yp, const float* res, int Nr, int nin,
                       const BlockP& bp){
    cemlp_block_kernel<<<Nr/16, 256, 0, stream>>>(Xp, Yp, res, Nr, nin,
        bp.lin_w, bp.lin_b, bp.silu_a, bp.silu_b, bp.lr_w, bp.norm_a,
        bp.gp_w, bp.ll_w, bp.ll_b, bp.ln_a);
  };

  // ---- preprocessing ----
  copy4_kernel<<<cdiv((long)N*3/4,256),256,0,stream>>>((float4*)pos_c, (const float4*)pos, (long)N*3/4);
  center_kernel<<<B,256,0,stream>>>(pos_c, idx0, N0G);
  natt_kernel<<<cdiv((long)N*3,256),256,0,stream>>>(natt, node_types, sim_emb, N);

  // ---- simplicial embedding ----
  embed0_kernel<<<cdiv((long)N0*16,256),256,0,stream>>>(xA, pos_c, vel, idx0,
      x_ind, x_ind_ptr, x_ind_batch, cl0_w, cl0_b, N0);

  float* F1 = m_in; float* O1 = msgA;                        // alias scratch
  feats1_kernel<<<cdiv((long)N1*8,256),256,0,stream>>>(F1, pos_c, vel, idx1,
      x_ind, x_ind_ptr, x_ind_batch, N1);
  run_block(F1, O1, nullptr, N1*2, 4, cl1);
  sum_perms_kernel<<<cdiv((long)N1*32,256),256,0,stream>>>(xA, O1, idx1, N1, 2);

  float* F2 = m_in; float* T2 = msgA; float* O2 = msgB;      // alias scratch
  feats2_kernel<<<cdiv((long)N2*36,256),256,0,stream>>>(F2, pos_c, vel, idx2,
      x_ind, x_ind_ptr, x_ind_batch, N2);
  run_block(F2, T2, nullptr, N2*6, 6,  cl2_0);
  run_block(T2, O2, nullptr, N2*6, 16, cl2_1);
  sum_perms_kernel<<<cdiv((long)N2*32,256),256,0,stream>>>(xA, O2, idx2, N2, 6);

  // ---- 4 EGCL layers ----
  float* xcur = xA; float* xnext = xB;
  for (int l=0;l<4;++l){
    edge_in_kernel<<<cdiv((long)E*22,256),256,0,stream>>>(m_in, xcur, natt, edge_index, E);
    run_block(m_in, msgA, nullptr, E, 22, ed[l][0]);
    run_block(msgA, msgB, nullptr, E, 16, ed[l][1]);
    zero4_kernel<<<cdiv((long)N*32,256),256,0,stream>>>((float4*)agg, (long)N*32);
    zero4_kernel<<<cdiv((long)N/4,256),256,0,stream>>>((float4*)cnt, (long)N/4);
    cnt_kernel<<<cdiv((long)E,256),256,0,stream>>>(cnt, edge_index, E);
    scatter_kernel<<<cdiv((long)E*128,256),256,0,stream>>>(agg, msgB, edge_index, E);
    aggnorm_kernel<<<cdiv((long)N*32,256),256,0,stream>>>(agg, cnt, N);
    node_in_kernel<<<cdiv((long)N*35,256),256,0,stream>>>(ninb, xcur, agg, natt, N);
    run_block(ninb, msgA, nullptr, N, 35, nd[l][0]);         // msgA reused as tmp
    run_block(msgA, xnext, xcur, N, 16, nd[l][1]);           // + residual h
    float* t = xcur; xcur = xnext; xnext = t;
  }

  // ---- projection + loss ----
  float* projin = msgA; float* projout = msgB;
  gather_nodes_kernel<<<cdiv((long)N0*32,256),256,0,stream>>>(projin, xcur, idx0, N0);
  run_block(projin, projout, nullptr, N0, 16, pm);
  predloss_kernel<<<cdiv((long)N0,256),256,0,stream>>>((float*)d_out, projout, pl_w,
      pos_c, idx0, yref, N0);
  lossmean_kernel<<<1,256,0,stream>>>((float*)d_out, N0);
}